// MultiLevelHybridHashEncoding_43928925504250
// MI455X (gfx1250) — compile-verified
//
#include <hip/hip_runtime.h>
#include <hip/hip_bf16.h>
#include <stdint.h>

// ---------------------------------------------------------------------------
// Multi-level hash-grid encoding (instant-NGP style), CDNA5 / gfx1250.
// Strategy: one block per (level, point-chunk). The level's embedding table
// (<=128KB) is staged into LDS with a single Tensor Data Mover
// tensor_load_to_lds (TENSORcnt-tracked), then all gathers are ds_load_b64.
// ---------------------------------------------------------------------------

#define N_LEVELS        16
#define TABLE_ROWS      16384          // padded rows per level in emb
#define BLOCK_THREADS   512
#define BLOCKS_PER_LVL  64
#define LDS_BYTES       131072         // 16384 * 2 * 4

// floor(16 * 2^(l/3)) in float32, replayed exactly from the reference.
__device__ __constant__ int c_res[N_LEVELS] =
    {16, 20, 25, 32, 40, 50, 64, 80, 101, 128, 161, 203, 256, 322, 406, 512};
// min(res^3, 16384)
__device__ __constant__ int c_nenc[N_LEVELS] =
    {4096, 8000, 15625, 16384, 16384, 16384, 16384, 16384,
     16384, 16384, 16384, 16384, 16384, 16384, 16384, 16384};

typedef uint32_t u32x4 __attribute__((ext_vector_type(4)));
typedef uint32_t u32x8 __attribute__((ext_vector_type(8)));

template <bool HASH>
__global__ __launch_bounds__(BLOCK_THREADS)
void hashenc_kernel(const float* __restrict__ x,
                    const float* __restrict__ emb,
                    float2* __restrict__ out,
                    int level_base, int npts, int chunk)
{
    extern __shared__ float2 table[];   // up to 16384 x float2 = 128 KB

    const int lvl   = level_base + (int)blockIdx.y;
    const int res   = c_res[lvl];
    const int n_enc = c_nenc[lvl];
    const uint32_t nd = (uint32_t)(n_enc * 2);   // dwords to DMA

    // ------------------------------------------------------------------
    // Stage this level's table into LDS via the Tensor Data Mover.
    // D# group0: count=1 | lds_addr | 57-bit global_addr | type=2
    // D# group1: data_size=4B, tensor_dim0=tile_dim0=nd (1-D tile)
    // ------------------------------------------------------------------
    if (threadIdx.x == 0) {
        uint64_t gaddr = (uint64_t)(uintptr_t)(emb + (size_t)lvl * (TABLE_ROWS * 2));
        uint32_t laddr = (uint32_t)(uintptr_t)table;   // low 32 bits = LDS offset
        u32x4 g0;
        g0[0] = 1u;                                             // count=1, user mode
        g0[1] = laddr;                                          // lds_addr
        g0[2] = (uint32_t)gaddr;                                // global_addr[31:0]
        g0[3] = ((uint32_t)(gaddr >> 32) & 0x01FFFFFFu)         // global_addr[56:32]
                | (2u << 30);                                   // type=2 ("image")
        u32x8 g1;
        g1[0] = 2u << 16;                                       // data_size=2 (4B), no mask
        g1[1] = (nd & 0xFFFFu) << 16;                           // tensor_dim0[15:0]
        g1[2] = (nd >> 16) & 0xFFFFu;                           // tensor_dim0[31:16], dim1=0
        g1[3] = (nd & 0xFFFFu) << 16;                           // tile_dim0 (<=32768)
        g1[4] = 0u;                                             // tile_dim1/2 unused
        g1[5] = nd;                                             // tensor_dim0_stride[31:0]
        g1[6] = 0u;
        g1[7] = 0u;
        asm volatile("tensor_load_to_lds %0, %1" :: "s"(g0), "s"(g1) : "memory");
    }
#if __has_builtin(__builtin_amdgcn_s_wait_tensorcnt)
    __builtin_amdgcn_s_wait_tensorcnt(0);
#else
    asm volatile("s_wait_tensorcnt 0x0" ::: "memory");
#endif
    __syncthreads();

    // ------------------------------------------------------------------
    // Per-point trilinear hash-grid interpolation, table served from LDS.
    // ------------------------------------------------------------------
    const float half_res = 0.5f * (float)res;   // exact: res*0.5
    const uint32_t ures  = (uint32_t)res;
    const int start = (int)blockIdx.x * chunk;

    for (int i = (int)threadIdx.x; i < chunk; i += BLOCK_THREADS) {
        const int b = start + i;
        if (b >= npts) break;
        const float* xb = x + 3 * (size_t)b;
        // xs = (x + 1) * res * 0.5 - 0.5  (power-of-two scaling: bit-exact regroup)
        float xs0 = (xb[0] + 1.0f) * half_res - 0.5f;
        float xs1 = (xb[1] + 1.0f) * half_res - 0.5f;
        float xs2 = (xb[2] + 1.0f) * half_res - 0.5f;
        float fl0 = floorf(xs0), fl1 = floorf(xs1), fl2 = floorf(xs2);
        float f0 = xs0 - fl0, f1 = xs1 - fl1, f2 = xs2 - fl2;
        int c0 = (int)fl0, c1 = (int)fl1, c2 = (int)fl2;

        // per-dim validity for offset 0 / offset 1 (unsigned compare = 0<=c<res)
        bool v0a = (uint32_t)c0 < ures,       v0b = (uint32_t)(c0 + 1) < ures;
        bool v1a = (uint32_t)c1 < ures,       v1b = (uint32_t)(c1 + 1) < ures;
        bool v2a = (uint32_t)c2 < ures,       v2b = (uint32_t)(c2 + 1) < ures;
        // per-dim weights
        float w0a = 1.0f - f0, w0b = f0;
        float w1a = 1.0f - f1, w1b = f1;
        float w2a = 1.0f - f2, w2b = f2;
        // per-dim index partials
        uint32_t h0a, h0b, h1a, h1b, h2a, h2b;
        if (HASH) {
            h0a = (uint32_t)c0;                     h0b = (uint32_t)(c0 + 1);
            h1a = (uint32_t)c1 * 2654435761u;       h1b = (uint32_t)(c1 + 1) * 2654435761u;
            h2a = (uint32_t)c2 * 805459861u;        h2b = (uint32_t)(c2 + 1) * 805459861u;
        } else {
            h0a = (uint32_t)c0;                     h0b = (uint32_t)(c0 + 1);
            h1a = (uint32_t)(c1 * res);             h1b = (uint32_t)((c1 + 1) * res);
            h2a = (uint32_t)(c2 * res * res);       h2b = (uint32_t)((c2 + 1) * res * res);
        }

        float acc0 = 0.0f, acc1 = 0.0f;
#define CORNER(va, vb, vc, ha, hb, hc, wa, wb, wc)                         \
        {                                                                  \
            bool v = (va) && (vb) && (vc);                                 \
            uint32_t idx;                                                  \
            if (HASH) idx = ((ha) ^ (hb) ^ (hc)) & 16383u;                 \
            else      idx = (ha) + (hb) + (hc);                            \
            idx = v ? idx : 0u;                                            \
            float w = v ? (wa) * (wb) * (wc) : 0.0f;                       \
            float2 e = table[idx];                                         \
            acc0 = fmaf(w, e.x, acc0);                                     \
            acc1 = fmaf(w, e.y, acc1);                                     \
        }
        // corner order k: offsets = [(k>>2)&1, (k>>1)&1, k&1]
        CORNER(v0a, v1a, v2a, h0a, h1a, h2a, w0a, w1a, w2a)   // 000
        CORNER(v0a, v1a, v2b, h0a, h1a, h2b, w0a, w1a, w2b)   // 001
        CORNER(v0a, v1b, v2a, h0a, h1b, h2a, w0a, w1b, w2a)   // 010
        CORNER(v0a, v1b, v2b, h0a, h1b, h2b, w0a, w1b, w2b)   // 011
        CORNER(v0b, v1a, v2a, h0b, h1a, h2a, w0b, w1a, w2a)   // 100
        CORNER(v0b, v1a, v2b, h0b, h1a, h2b, w0b, w1a, w2b)   // 101
        CORNER(v0b, v1b, v2a, h0b, h1b, h2a, w0b, w1b, w2a)   // 110
        CORNER(v0b, v1b, v2b, h0b, h1b, h2b, w0b, w1b, w2b)   // 111
#undef CORNER

        out[(size_t)b * N_LEVELS + lvl] = make_float2(acc0, acc1);
    }
}

extern "C" void kernel_launch(void* const* d_in, const int* in_sizes, int n_in,
                              void* d_out, int out_size, void* d_ws, size_t ws_size,
                              hipStream_t stream)
{
    const float* x   = (const float*)d_in[0];
    const float* emb = (const float*)d_in[1];
    float2* out      = (float2*)d_out;

    const int npts  = in_sizes[0] / 3;
    const int chunk = (npts + BLOCKS_PER_LVL - 1) / BLOCKS_PER_LVL;

    // Opt in to >64KB dynamic LDS (CDNA5 WGP has 320KB).
    (void)hipFuncSetAttribute(reinterpret_cast<const void*>(&hashenc_kernel<false>),
                              hipFuncAttributeMaxDynamicSharedMemorySize, LDS_BYTES);
    (void)hipFuncSetAttribute(reinterpret_cast<const void*>(&hashenc_kernel<true>),
                              hipFuncAttributeMaxDynamicSharedMemorySize, LDS_BYTES);

    dim3 block(BLOCK_THREADS);
    // Levels 0..2: direct ravel indexing (res^3 <= 16384).
    hashenc_kernel<false><<<dim3(BLOCKS_PER_LVL, 3), block, LDS_BYTES, stream>>>(
        x, emb, out, 0, npts, chunk);
    // Levels 3..15: xor-hash indexing into 16384-entry tables.
    hashenc_kernel<true><<<dim3(BLOCKS_PER_LVL, 13), block, LDS_BYTES, stream>>>(
        x, emb, out, 3, npts, chunk);
}